// HyperGraph_7447473291952
// MI455X (gfx1250) — compile-verified
//
#include <hip/hip_runtime.h>

#define N_NODES 100000
#define N_HEDGES 20000
#define N_INC 800000
#define F_IN 128
#define HEADS 4
#define OUT1 32
#define F1 128          // HEADS*OUT1
#define OUT2 16
#define NEG_SLOPE 0.2f

typedef float v2f __attribute__((ext_vector_type(2)));
typedef float v8f __attribute__((ext_vector_type(8)));

// ---------------- utility kernels ----------------

__global__ void zero_f32_kernel(float* __restrict__ p, long n) {
  long i = (long)blockIdx.x * blockDim.x + threadIdx.x;
  if (i < n) p[i] = 0.f;
}

__global__ void degree_kernel(const int* __restrict__ ni, const int* __restrict__ hi,
                              float* __restrict__ dn, float* __restrict__ dh) {
  int i = blockIdx.x * blockDim.x + threadIdx.x;
  if (i < N_INC) {
    atomicAdd(&dn[ni[i]], 1.f);
    atomicAdd(&dh[hi[i]], 1.f);
  }
}

__global__ void invdeg_kernel(float* __restrict__ dn, float* __restrict__ dh) {
  int i = blockIdx.x * blockDim.x + threadIdx.x;
  if (i < N_NODES) {
    float d = dn[i];
    dn[i] = d > 0.f ? 1.f / d : 0.f;
  } else if (i < N_NODES + N_HEDGES) {
    int j = i - N_NODES;
    float d = dh[j];
    dh[j] = d > 0.f ? 1.f / d : 0.f;
  }
}

// ---------------- fp32 WMMA GEMM: C = A(MxK) * B(KxN) ----------------
// One wave per 16x16 C tile. V_WMMA_F32_16X16X4_F32 layouts (ISA 7.12.2):
//   A 16x4 : lanes 0-15 -> rows M=0..15, VGPR{0,1} = K{0,1}; lanes 16-31 = K{2,3}
//   B 4x16 : lanes 0-15 -> cols N=0..15, VGPR{0,1} = K{0,1}; lanes 16-31 = K{2,3}
//   C 16x16: VGPR r: lanes 0-15 = row r, lanes 16-31 = row 8+r
// M, N multiples of 16; K multiple of 4.
__global__ void wmma_gemm_f32_kernel(const float* __restrict__ A, const float* __restrict__ B,
                                     float* __restrict__ C, int M, int N, int K) {
  int lane = threadIdx.x & 31;
  int wave = threadIdx.x >> 5;
  int ntn = N >> 4;
  int total = (M >> 4) * ntn;
  int tile = blockIdx.x * 8 + wave;
  if (tile >= total) return;
  int tm = tile / ntn, tn = tile - tm * ntn;
  int m0 = tm << 4, n0 = tn << 4;
  int half = lane >> 4, l = lane & 15;

  v8f acc = {};
  const float* arow = A + (size_t)(m0 + l) * K + 2 * half;         // two consecutive K
  const float* bcol = B + (size_t)(2 * half) * N + n0 + l;         // two K rows, stride N
  for (int k0 = 0; k0 < K; k0 += 4) {
    v2f a; a.x = arow[0]; a.y = arow[1];
    v2f b; b.x = bcol[0]; b.y = bcol[N];
    acc = __builtin_amdgcn_wmma_f32_16x16x4_f32(
        /*neg_a=*/false, a, /*neg_b=*/false, b,
        /*c_mod=*/(short)0, acc, /*reuse_a=*/false, /*reuse_b=*/false);
    arow += 4;
    bcol += (size_t)4 * N;
  }
  float* crow = C + (size_t)(m0 + half * 8) * N + n0 + l;
#pragma unroll
  for (int r = 0; r < 8; ++r) crow[(size_t)r * N] = acc[r];
}

// ---------------- attention ----------------

// scores: out[row,h] = dot(fw[row, h*32 : h*32+32], att[h, attOff : attOff+32])
__global__ void score_kernel(const float* __restrict__ fw, const float* __restrict__ att,
                             float* __restrict__ out, int rows, int attOff) {
  int idx = blockIdx.x * blockDim.x + threadIdx.x;
  if (idx >= rows * HEADS) return;
  int r = idx >> 2, h = idx & 3;
  const float* f = fw + (size_t)r * F1 + h * OUT1;
  const float* a = att + h * (2 * OUT1) + attOff;
  float s = 0.f;
#pragma unroll
  for (int c = 0; c < OUT1; ++c) s += f[c] * a[c];
  out[idx] = s;
}

// order-preserving float<->uint encoding for atomic max
__device__ __forceinline__ unsigned enc_ord(float f) {
  unsigned u = __float_as_uint(f);
  return (u & 0x80000000u) ? ~u : (u | 0x80000000u);
}
__device__ __forceinline__ float dec_ord(unsigned e) {
  return (e & 0x80000000u) ? __uint_as_float(e & 0x7FFFFFFFu) : __uint_as_float(~e);
}

__global__ void logit_max_kernel(const float* __restrict__ sn, const float* __restrict__ se,
                                 const int* __restrict__ ni, const int* __restrict__ hi,
                                 float* __restrict__ alpha, unsigned* __restrict__ maxenc) {
  int idx = blockIdx.x * blockDim.x + threadIdx.x;
  if (idx >= N_INC * HEADS) return;
  int i = idx >> 2, h = idx & 3;
  float a = sn[(size_t)ni[i] * HEADS + h] + se[(size_t)hi[i] * HEADS + h];
  a = a > 0.f ? a : NEG_SLOPE * a;              // leaky relu
  alpha[idx] = a;
  atomicMax(&maxenc[(size_t)hi[i] * HEADS + h], enc_ord(a));
}

__global__ void exp_sum_kernel(const int* __restrict__ hi, float* __restrict__ alpha,
                               const unsigned* __restrict__ maxenc, float* __restrict__ hsum) {
  int idx = blockIdx.x * blockDim.x + threadIdx.x;
  if (idx >= N_INC * HEADS) return;
  int i = idx >> 2, h = idx & 3;
  unsigned e = maxenc[(size_t)hi[i] * HEADS + h];
  float amax = (e == 0u) ? 0.f : dec_ord(e);
  if (!__builtin_isfinite(amax)) amax = 0.f;
  float v = __expf(alpha[idx] - amax);
  alpha[idx] = v;
  atomicAdd(&hsum[(size_t)hi[i] * HEADS + h], v);
}

__global__ void norm_kernel(const int* __restrict__ hi, float* __restrict__ alpha,
                            const float* __restrict__ hsum) {
  int idx = blockIdx.x * blockDim.x + threadIdx.x;
  if (idx >= N_INC * HEADS) return;
  int i = idx >> 2, h = idx & 3;
  alpha[idx] = alpha[idx] / (hsum[(size_t)hi[i] * HEADS + h] + 1e-16f);
}

// ---------------- layer-1 scatter stages (one wave per incidence, lane = channel-in-head) ----

__global__ void agg1_kernel(const int* __restrict__ ni, const int* __restrict__ hi,
                            const float* __restrict__ alpha, const float* __restrict__ dh,
                            const float* __restrict__ xw, float* __restrict__ eout1) {
  int lane = threadIdx.x & 31;
  int i = blockIdx.x * (blockDim.x >> 5) + (threadIdx.x >> 5);
  if (i >= N_INC) return;
  int node = ni[i], he = hi[i];
  float Bv = dh[he];
#pragma unroll
  for (int h = 0; h < HEADS; ++h) {
    float w = Bv * alpha[(size_t)i * HEADS + h];
    int ch = h * OUT1 + lane;
    atomicAdd(&eout1[(size_t)he * F1 + ch], w * xw[(size_t)node * F1 + ch]);
  }
}

__global__ void agg2_kernel(const int* __restrict__ ni, const int* __restrict__ hi,
                            const float* __restrict__ alpha, const float* __restrict__ dn,
                            const float* __restrict__ eout1, float* __restrict__ hbuf) {
  int lane = threadIdx.x & 31;
  int i = blockIdx.x * (blockDim.x >> 5) + (threadIdx.x >> 5);
  if (i >= N_INC) return;
  int node = ni[i], he = hi[i];
  float Dv = dn[node];
#pragma unroll
  for (int h = 0; h < HEADS; ++h) {
    float w = Dv * alpha[(size_t)i * HEADS + h];
    int ch = h * OUT1 + lane;
    atomicAdd(&hbuf[(size_t)node * F1 + ch], w * eout1[(size_t)he * F1 + ch]);
  }
}

__global__ void bias_relu_kernel(float* __restrict__ hbuf, const float* __restrict__ b1) {
  long i = (long)blockIdx.x * blockDim.x + threadIdx.x;
  if (i < (long)N_NODES * F1) {
    float v = hbuf[i] + b1[i & (F1 - 1)];
    hbuf[i] = v > 0.f ? v : 0.f;
  }
}

// ---------------- layer-2 aggregation ----------------

__global__ void agg3_kernel(const int* __restrict__ ni, const int* __restrict__ hi,
                            const float* __restrict__ dh, const float* __restrict__ h2w,
                            float* __restrict__ eout2) {
  long idx = (long)blockIdx.x * blockDim.x + threadIdx.x;
  if (idx >= (long)N_INC * OUT2) return;
  int i = (int)(idx >> 4), c = (int)(idx & 15);
  int node = ni[i], he = hi[i];
  atomicAdd(&eout2[(size_t)he * OUT2 + c], dh[he] * h2w[(size_t)node * OUT2 + c]);
}

// final: out[c] = sum_i D[node_i]*eout2[hedge_i,c] + N_NODES*b2[c]
__global__ void final_kernel(const int* __restrict__ ni, const int* __restrict__ hi,
                             const float* __restrict__ dn, const float* __restrict__ eout2,
                             const float* __restrict__ b2, float* __restrict__ out) {
  __shared__ float part[OUT2];
  int t = threadIdx.x;
  if (t < OUT2) part[t] = 0.f;
  __syncthreads();
  int c = t & 15;
  long groups = (long)gridDim.x * blockDim.x / OUT2;
  long g = ((long)blockIdx.x * blockDim.x + t) >> 4;
  float acc = 0.f;
  for (long i = g; i < N_INC; i += groups)
    acc += dn[ni[i]] * eout2[(size_t)hi[i] * OUT2 + c];
  atomicAdd(&part[c], acc);
  __syncthreads();
  if (t < OUT2) {
    float v = part[t];
    if (blockIdx.x == 0) v += (float)N_NODES * b2[t];
    atomicAdd(&out[t], v);
  }
}

// ---------------- launch ----------------

static inline int cdiv(long a, long b) { return (int)((a + b - 1) / b); }

extern "C" void kernel_launch(void* const* d_in, const int* in_sizes, int n_in,
                              void* d_out, int out_size, void* d_ws, size_t ws_size,
                              hipStream_t stream) {
  const float* x          = (const float*)d_in[0];
  const float* hyper_attr = (const float*)d_in[1];
  const float* W1         = (const float*)d_in[2];
  const float* att1       = (const float*)d_in[3];
  const float* b1         = (const float*)d_in[4];
  const float* W2         = (const float*)d_in[5];
  const float* b2         = (const float*)d_in[6];
  const int*   node_idx   = (const int*)d_in[7];
  const int*   hedge_idx  = (const int*)d_in[8];
  float* out = (float*)d_out;

  float* ws = (float*)d_ws;
  size_t off = 0;
  float* xw1    = ws + off; off += (size_t)N_NODES * F1;    // reused later as hbuf (layer-1 output)
  float* ew1    = ws + off; off += (size_t)N_HEDGES * F1;
  float* h2w    = ws + off; off += (size_t)N_NODES * OUT2;
  float* sn     = ws + off; off += (size_t)N_NODES * HEADS;
  float* se     = ws + off; off += (size_t)N_HEDGES * HEADS;
  float* alpha  = ws + off; off += (size_t)N_INC * HEADS;
  float* zbase  = ws + off;                                  // zero-init region starts here
  float* dn     = ws + off; off += N_NODES;
  float* dh     = ws + off; off += N_HEDGES;
  unsigned* maxenc = (unsigned*)(ws + off); off += (size_t)N_HEDGES * HEADS;
  float* hsum   = ws + off; off += (size_t)N_HEDGES * HEADS;
  float* eout1  = ws + off; off += (size_t)N_HEDGES * F1;
  float* eout2  = ws + off; off += (size_t)N_HEDGES * OUT2;
  long nzero = (long)((ws + off) - zbase);

  const int B = 256;

  // init accumulators + output
  zero_f32_kernel<<<cdiv(nzero, B), B, 0, stream>>>(zbase, nzero);
  zero_f32_kernel<<<1, 32, 0, stream>>>(out, OUT2);

  // degrees -> inverse degrees
  degree_kernel<<<cdiv(N_INC, B), B, 0, stream>>>(node_idx, hedge_idx, dn, dh);
  invdeg_kernel<<<cdiv(N_NODES + N_HEDGES, B), B, 0, stream>>>(dn, dh);

  // dense GEMMs on the WMMA pipe (fp32)
  {
    int tiles = (N_NODES / 16) * (F1 / 16);
    wmma_gemm_f32_kernel<<<cdiv(tiles, 8), B, 0, stream>>>(x, W1, xw1, N_NODES, F1, F_IN);
  }
  {
    int tiles = (N_HEDGES / 16) * (F1 / 16);
    wmma_gemm_f32_kernel<<<cdiv(tiles, 8), B, 0, stream>>>(hyper_attr, W1, ew1, N_HEDGES, F1, F_IN);
  }

  // attention scores, segment softmax over hyperedges
  score_kernel<<<cdiv((long)N_NODES * HEADS, B), B, 0, stream>>>(xw1, att1, sn, N_NODES, 0);
  score_kernel<<<cdiv((long)N_HEDGES * HEADS, B), B, 0, stream>>>(ew1, att1, se, N_HEDGES, OUT1);
  logit_max_kernel<<<cdiv((long)N_INC * HEADS, B), B, 0, stream>>>(sn, se, node_idx, hedge_idx, alpha, maxenc);
  exp_sum_kernel<<<cdiv((long)N_INC * HEADS, B), B, 0, stream>>>(hedge_idx, alpha, maxenc, hsum);
  norm_kernel<<<cdiv((long)N_INC * HEADS, B), B, 0, stream>>>(hedge_idx, alpha, hsum);

  // layer-1 two-stage aggregation
  agg1_kernel<<<cdiv(N_INC, B / 32), B, 0, stream>>>(node_idx, hedge_idx, alpha, dh, xw1, eout1);
  // xw1 is no longer needed -> reuse as layer-1 output buffer (zero first)
  float* hbuf = xw1;
  zero_f32_kernel<<<cdiv((long)N_NODES * F1, B), B, 0, stream>>>(hbuf, (long)N_NODES * F1);
  agg2_kernel<<<cdiv(N_INC, B / 32), B, 0, stream>>>(node_idx, hedge_idx, alpha, dn, eout1, hbuf);
  bias_relu_kernel<<<cdiv((long)N_NODES * F1, B), B, 0, stream>>>(hbuf, b1);

  // layer 2: GEMM + aggregation + global reduction
  {
    int tiles = (N_NODES / 16) * (OUT2 / 16);
    wmma_gemm_f32_kernel<<<cdiv(tiles, 8), B, 0, stream>>>(hbuf, W2, h2w, N_NODES, OUT2, F1);
  }
  agg3_kernel<<<cdiv((long)N_INC * OUT2, B), B, 0, stream>>>(node_idx, hedge_idx, dh, h2w, eout2);
  final_kernel<<<1024, B, 0, stream>>>(node_idx, hedge_idx, dn, eout2, b2, out);
}